// StreamingMSTCN_Export_33792802684995
// MI455X (gfx1250) — compile-verified
//
#include <hip/hip_runtime.h>
#include <math.h>

// ---------------------------------------------------------------------------
// StreamingMSTCN for MI455X (gfx1250, wave32).
// Memory-bound on buffer shifts (~0.95 GB traffic -> ~41us @ 23.3 TB/s);
// GEMMs (~3 GFLOP) done with V_WMMA_F32_16X16X4_F32 (exact fp32 path).
// ---------------------------------------------------------------------------

typedef float v2f __attribute__((ext_vector_type(2)));
typedef float v8f __attribute__((ext_vector_type(8)));

#define B_SZ 2048
#define C_CH 320
#define F_CH 320

// ---------------------------------------------------------------------------
// Buffer shift: news[b,l,:] = (l < L-1) ? buf[b,l+1,:] : x[b,0,:]
// Pure bandwidth: float4 (global_*_b128) streaming.
// ---------------------------------------------------------------------------
__global__ __launch_bounds__(256)
void shift_copy_kernel(const float4* __restrict__ buf,
                       const float4* __restrict__ x,
                       float4* __restrict__ dst,
                       int L, long total)
{
    long t = (long)blockIdx.x * blockDim.x + threadIdx.x;
    if (t >= total) return;
    const int QPR = C_CH / 4;               // 80 float4 per channel row
    int  q     = (int)(t % QPR);
    long rowid = t / QPR;                   // b*L + l
    int  l     = (int)(rowid % L);
    long b     = rowid / L;
    const float4* src = (l < L - 1) ? (buf + (b * L + l + 1) * QPR)
                                    : (x   + b * QPR);
    dst[rowid * QPR + q] = src[q];
}

// ---------------------------------------------------------------------------
// Wave-level fp32 WMMA tile helpers.
// V_WMMA_F32_16X16X4_F32 layouts (ISA 7.12.2):
//   A 16x4 : lane<16 -> row=lane, {K0,K1}; lane>=16 -> row=lane-16, {K2,K3}
//   B 4x16 : lane<16 -> col=lane, {K0,K1}; lane>=16 -> col=lane-16, {K2,K3}
//   C 16x16: VGPR v  : lane<16 -> M=v, N=lane ; lane>=16 -> M=v+8, N=lane-16
// ---------------------------------------------------------------------------

// ---------------------------------------------------------------------------
// Branch GEMMs -> merged (2048 x 320), ReLU epilogue.
// One wave computes a 16x16 tile. ntile (0..19) selects branch = ntile/4.
// taps[b, j, :] = (j < k-1) ? buf[b, j*dil, :] : x[b, :]
// ---------------------------------------------------------------------------
__global__ __launch_bounds__(256)
void branch_gemm_kernel(const float* __restrict__ x,
                        const float* __restrict__ buf1, const float* __restrict__ W1, const float* __restrict__ bb1,
                        const float* __restrict__ buf2, const float* __restrict__ W2, const float* __restrict__ bb2,
                        const float* __restrict__ buf3, const float* __restrict__ W3, const float* __restrict__ bb3,
                        const float* __restrict__ buf4, const float* __restrict__ W4, const float* __restrict__ bb4,
                        const float* __restrict__ buf5, const float* __restrict__ W5, const float* __restrict__ bb5,
                        float* __restrict__ merged)
{
    const int lane = threadIdx.x & 31;
    const int wave = threadIdx.x >> 5;
    const int w    = blockIdx.x * 8 + wave;      // 2560 tiles total
    const int ntile = w % 20;                    // 20 tiles of 16 cols = 320
    const int mtile = w / 20;                    // 128 row tiles
    const int branch = ntile >> 2;               // 4 tiles per 64-col branch
    const int colseg = (ntile & 3) * 16;         // col offset within branch

    // wave-uniform branch parameter selection
    const float* buf; const float* W; const float* bias;
    int k, dil, L;
    if      (branch == 0) { buf = buf1; W = W1; bias = bb1; k = 3; dil = 1;  L = 2;  }
    else if (branch == 1) { buf = buf2; W = W2; bias = bb2; k = 3; dil = 2;  L = 4;  }
    else if (branch == 2) { buf = buf3; W = W3; bias = bb3; k = 5; dil = 4;  L = 16; }
    else if (branch == 3) { buf = buf4; W = W4; bias = bb4; k = 9; dil = 8;  L = 64; }
    else                  { buf = buf5; W = W5; bias = bb5; k = 9; dil = 12; L = 96; }

    const int half = lane >> 4;
    const int r    = lane & 15;
    const int row  = mtile * 16 + r;

    v8f acc = {};
    for (int j = 0; j < k; ++j) {
        const float* arow = (j < k - 1)
            ? (buf + ((size_t)row * L + (size_t)j * dil) * C_CH)
            : (x   + (size_t)row * C_CH);
        const float* wbase = W + (size_t)j * C_CH * 64;   // W[j][c][o], cout=64
        for (int c = 0; c < C_CH; c += 4) {
            const int k0 = c + 2 * half;
            v2f a;
            a.x = arow[k0];
            a.y = arow[k0 + 1];
            v2f bf;
            bf.x = wbase[(size_t)k0       * 64 + colseg + r];
            bf.y = wbase[(size_t)(k0 + 1) * 64 + colseg + r];
            acc = __builtin_amdgcn_wmma_f32_16x16x4_f32(
                false, a, false, bf, (short)0, acc, false, false);
        }
    }

    const int   gcol  = branch * 64 + colseg + r;
    const float bv    = bias[colseg + r];
    const int   rbase = mtile * 16 + half * 8;
    #pragma unroll
    for (int v = 0; v < 8; ++v) {
        float val = acc[v] + bv;
        val = fmaxf(val, 0.0f);                       // ReLU
        merged[(size_t)(rbase + v) * F_CH + gcol] = val;
    }
}

// ---------------------------------------------------------------------------
// Generic WMMA fp32 GEMM: D = act(A (*Amul) @ Bm + bias)
// ACT: 0=ReLU, 1=sigmoid, 2=none. HAS_MUL: elementwise A multiplier.
// One wave per 16x16 tile; grid must cover M/16 * N/16 tiles (8 waves/block).
// ---------------------------------------------------------------------------
template <int ACT, bool HAS_MUL>
__global__ __launch_bounds__(256)
void gemm_wmma_kernel(const float* __restrict__ A,
                      const float* __restrict__ Amul,
                      const float* __restrict__ Bm,
                      const float* __restrict__ bias,
                      float* __restrict__ D,
                      int M, int N, int K)
{
    const int lane  = threadIdx.x & 31;
    const int wave  = threadIdx.x >> 5;
    const int w     = blockIdx.x * 8 + wave;
    const int ntcnt = N / 16;
    const int ntile = w % ntcnt;
    const int mtile = w / ntcnt;

    const int half = lane >> 4;
    const int r    = lane & 15;
    const int row  = mtile * 16 + r;

    const float* arow  = A + (size_t)row * K;
    const float* amrow = HAS_MUL ? (Amul + (size_t)row * K) : nullptr;

    v8f acc = {};
    for (int c = 0; c < K; c += 4) {
        const int k0 = c + 2 * half;
        v2f a;
        a.x = arow[k0];
        a.y = arow[k0 + 1];
        if (HAS_MUL) {
            a.x *= amrow[k0];
            a.y *= amrow[k0 + 1];
        }
        v2f bf;
        bf.x = Bm[(size_t)k0       * N + ntile * 16 + r];
        bf.y = Bm[(size_t)(k0 + 1) * N + ntile * 16 + r];
        acc = __builtin_amdgcn_wmma_f32_16x16x4_f32(
            false, a, false, bf, (short)0, acc, false, false);
    }

    const int   gcol  = ntile * 16 + r;
    const float bv    = bias[gcol];
    const int   rbase = mtile * 16 + half * 8;
    #pragma unroll
    for (int v = 0; v < 8; ++v) {
        float val = acc[v] + bv;
        if (ACT == 0)      val = fmaxf(val, 0.0f);
        else if (ACT == 1) val = 1.0f / (1.0f + __expf(-val));
        D[(size_t)(rbase + v) * N + gcol] = val;
    }
}

// ---------------------------------------------------------------------------
extern "C" void kernel_launch(void* const* d_in, const int* in_sizes, int n_in,
                              void* d_out, int out_size, void* d_ws, size_t ws_size,
                              hipStream_t stream)
{
    (void)in_sizes; (void)n_in; (void)out_size; (void)ws_size;

    const float* x = (const float*)d_in[0];
    const float* buf[5]; const float* W[5]; const float* bb[5];
    for (int i = 0; i < 5; ++i) {
        buf[i] = (const float*)d_in[1 + 3 * i];
        W[i]   = (const float*)d_in[2 + 3 * i];
        bb[i]  = (const float*)d_in[3 + 3 * i];
    }
    const float* Wa1  = (const float*)d_in[16];
    const float* ba1  = (const float*)d_in[17];
    const float* Wa2  = (const float*)d_in[18];
    const float* ba2  = (const float*)d_in[19];
    const float* Wout = (const float*)d_in[20];
    const float* bout = (const float*)d_in[21];

    float* out    = (float*)d_out;
    float* merged = (float*)d_ws;                          // 2048*320
    float* att1   = merged + (size_t)B_SZ * F_CH;          // 2048*80
    float* att2   = att1   + (size_t)B_SZ * (F_CH / 4);    // 2048*320

    // ---- buffer shifts (bandwidth-bound bulk of the work) ----
    const int Ls[5] = {2, 4, 16, 64, 96};
    size_t off = (size_t)B_SZ * F_CH;                      // after final_out
    for (int i = 0; i < 5; ++i) {
        long total  = (long)B_SZ * Ls[i] * (C_CH / 4);
        int  blocks = (int)((total + 255) / 256);
        shift_copy_kernel<<<blocks, 256, 0, stream>>>(
            (const float4*)buf[i], (const float4*)x,
            (float4*)(out + off), Ls[i], total);
        off += (size_t)B_SZ * Ls[i] * C_CH;
    }

    // ---- branch GEMMs -> merged (ReLU) : 2560 tiles / 8 waves = 320 blocks
    branch_gemm_kernel<<<320, 256, 0, stream>>>(
        x,
        buf[0], W[0], bb[0], buf[1], W[1], bb[1], buf[2], W[2], bb[2],
        buf[3], W[3], bb[3], buf[4], W[4], bb[4],
        merged);

    // ---- att1 = relu(merged @ Wa1 + ba1): (2048x320)(320x80) -> 640 tiles
    gemm_wmma_kernel<0, false><<<(128 * 5) / 8, 256, 0, stream>>>(
        merged, nullptr, Wa1, ba1, att1, B_SZ, 80, 320);

    // ---- att2 = sigmoid(att1 @ Wa2 + ba2): (2048x80)(80x320) -> 2560 tiles
    gemm_wmma_kernel<1, false><<<(128 * 20) / 8, 256, 0, stream>>>(
        att1, nullptr, Wa2, ba2, att2, B_SZ, 320, 80);

    // ---- final = (merged * att2) @ Wout + bout: (2048x320)(320x320)
    gemm_wmma_kernel<2, true><<<(128 * 20) / 8, 256, 0, stream>>>(
        merged, att2, Wout, bout, out, B_SZ, 320, 320);
}